// HungarianMatcher_67422396613141
// MI455X (gfx1250) — compile-verified
//
#include <hip/hip_runtime.h>
#include <stdint.h>

#define QN 1024   // queries per frame (blockDim.x of kernel 1)

__device__ __forceinline__ unsigned long long shfl_xor_u64(unsigned long long v, int mask) {
    unsigned lo = (unsigned)v;
    unsigned hi = (unsigned)(v >> 32);
    lo = __shfl_xor(lo, mask, 32);
    hi = __shfl_xor(hi, mask, 32);
    return ((unsigned long long)hi << 32) | (unsigned long long)lo;
}

// Monotonic float -> u32 mapping (order-preserving for all finite floats)
__device__ __forceinline__ unsigned float_key(float f) {
    unsigned u = __float_as_uint(f);
    return (u & 0x80000000u) ? ~u : (u ^ 0x80000000u);
}

__global__ __launch_bounds__(QN) void frame_cost_kernel(
    const float* __restrict__ logits_b,  // [F, Q, 2]
    const float* __restrict__ boxes,     // [F, Q, 4] (cx,cy,w,h)
    const float* __restrict__ tgt,       // [F, 4]    (cx,cy,w,h)
    float* __restrict__ outC,            // [F*Q]
    float* __restrict__ outRows,         // [F] (argmin index as float)
    int*   __restrict__ wsRows)          // [F] (argmin index as int, scratch)
{
    const int f = blockIdx.x;
    const int q = threadIdx.x;

    __shared__ float sBox[QN * 4];                 // 16 KB
    __shared__ float sLog[QN * 2];                 //  8 KB
    __shared__ unsigned long long sMin;

    // ---- CDNA5 async global->LDS staging (ASYNCcnt path) ----
    {
        const float* gB = boxes    + ((size_t)f * QN + q) * 4;
        const float* gL = logits_b + ((size_t)f * QN + q) * 2;
        unsigned ldsB = (unsigned)(uintptr_t)(&sBox[q * 4]);  // low 32 bits of flat = LDS offset
        unsigned ldsL = (unsigned)(uintptr_t)(&sLog[q * 2]);
        asm volatile("global_load_async_to_lds_b128 %0, %1, off"
                     :: "v"(ldsB), "v"(gB) : "memory");
        asm volatile("global_load_async_to_lds_b64 %0, %1, off"
                     :: "v"(ldsL), "v"(gL) : "memory");
    }
    if (q == 0) sMin = 0xFFFFFFFFFFFFFFFFull;
    asm volatile("s_wait_asynccnt 0" ::: "memory");   // drain this wave's async copies
    __syncthreads();

    // ---- per-query cost ----
    float bcx = sBox[q * 4 + 0], bcy = sBox[q * 4 + 1];
    float bw  = sBox[q * 4 + 2], bh  = sBox[q * 4 + 3];
    float l0  = sLog[q * 2 + 0], l1  = sLog[q * 2 + 1];

    // target box is uniform per block -> compiler scalarizes these loads
    float tcx = tgt[f * 4 + 0], tcy = tgt[f * 4 + 1];
    float tw  = tgt[f * 4 + 2], th  = tgt[f * 4 + 3];

    // L1 cost in cxcywh space
    float cost_bbox = fabsf(bcx - tcx) + fabsf(bcy - tcy) + fabsf(bw - tw) + fabsf(bh - th);

    // GIoU on xyxy boxes
    float px0 = bcx - 0.5f * bw, py0 = bcy - 0.5f * bh;
    float px1 = bcx + 0.5f * bw, py1 = bcy + 0.5f * bh;
    float tx0 = tcx - 0.5f * tw, ty0 = tcy - 0.5f * th;
    float tx1 = tcx + 0.5f * tw, ty1 = tcy + 0.5f * th;

    float area1 = (px1 - px0) * (py1 - py0);
    float area2 = (tx1 - tx0) * (ty1 - ty0);
    float iw = fmaxf(fminf(px1, tx1) - fmaxf(px0, tx0), 0.0f);
    float ih = fmaxf(fminf(py1, ty1) - fmaxf(py0, ty0), 0.0f);
    float inter = iw * ih;
    float uni   = area1 + area2 - inter;
    float iou   = inter / uni;
    float cw = fmaxf(fmaxf(px1, tx1) - fminf(px0, tx0), 0.0f);
    float ch = fmaxf(fmaxf(py1, ty1) - fminf(py0, ty0), 0.0f);
    float areac = cw * ch;
    float giou  = iou - (areac - uni) / areac;

    // -softmax(logits_b)[...,1] == -sigmoid(l1 - l0)
    float cost_class = -1.0f / (1.0f + expf(l0 - l1));

    float C = 5.0f * cost_bbox + 1.0f * cost_class - 2.0f * giou;

    outC[(size_t)f * QN + q] = C;

    // ---- argmin over Q: pack (ordered float key << 32 | index), wave32 reduce, ds_min_u64 ----
    unsigned long long packed =
        ((unsigned long long)float_key(C) << 32) | (unsigned long long)(unsigned)q;
    #pragma unroll
    for (int m = 16; m > 0; m >>= 1) {
        unsigned long long other = shfl_xor_u64(packed, m);
        packed = (other < packed) ? other : packed;
    }
    if ((q & 31) == 0) atomicMin(&sMin, packed);   // ds_min_u64
    __syncthreads();

    if (q == 0) {
        int row = (int)(sMin & 0xFFFFFFFFu);       // ties resolve to first index (argmin semantics)
        wsRows[f]  = row;
        outRows[f] = (float)row;
    }
}

__global__ __launch_bounds__(1024) void mode_kernel(
    const int* __restrict__ rows, int F, int Q, float* __restrict__ outMode)
{
    __shared__ int counts[QN];
    __shared__ unsigned long long sMax;

    int t = threadIdx.x;
    for (int b = t; b < QN; b += blockDim.x) counts[b] = 0;
    if (t == 0) sMax = 0;
    __syncthreads();

    for (int i = t; i < F; i += blockDim.x) atomicAdd(&counts[rows[i]], 1);
    __syncthreads();

    // argmax(bincount) with first-index tie-break: key = (count << 32) | (0x7FFFFFFF - bin)
    unsigned long long best = 0;
    for (int b = t; b < Q; b += blockDim.x) {
        unsigned long long key =
            ((unsigned long long)(unsigned)counts[b] << 32) |
            (unsigned long long)(0x7FFFFFFFu - (unsigned)b);
        if (key > best) best = key;
    }
    #pragma unroll
    for (int m = 16; m > 0; m >>= 1) {
        unsigned long long other = shfl_xor_u64(best, m);
        best = (other > best) ? other : best;
    }
    if ((t & 31) == 0) atomicMax(&sMax, best);     // ds_max_u64
    __syncthreads();

    if (t == 0) {
        unsigned bin = 0x7FFFFFFFu - (unsigned)(sMax & 0xFFFFFFFFu);
        outMode[0] = (float)bin;
    }
}

extern "C" void kernel_launch(void* const* d_in, const int* in_sizes, int n_in,
                              void* d_out, int out_size, void* d_ws, size_t ws_size,
                              hipStream_t stream) {
    // d_in[0]: pred_logits  [16,32,1024,80] f32  -- UNUSED by reference math (skip 168 MB)
    // d_in[1]: pred_logits_b [16,32,1024,2] f32
    // d_in[2]: pred_boxes    [16,32,1024,4] f32
    // d_in[3]: tgt_bbox      [16,32,1,4]    f32
    // d_in[4]: tgt_ids       [512]          i64  -- UNUSED
    const float* logits_b = (const float*)d_in[1];
    const float* boxes    = (const float*)d_in[2];
    const float* tgt      = (const float*)d_in[3];

    const int F = in_sizes[3] / 4;                 // bs*t = 512 frames
    const int Q = in_sizes[2] / (4 * F);           // 1024 queries

    float* out     = (float*)d_out;
    float* outC    = out;                          // F*Q cost values
    float* outRows = out + (size_t)F * Q;          // F argmin rows (as float)
    float* outMode = outRows + F;                  // 1 mode index

    int* wsRows = (int*)d_ws;                      // F ints of scratch

    frame_cost_kernel<<<F, Q, 0, stream>>>(logits_b, boxes, tgt, outC, outRows, wsRows);
    mode_kernel<<<1, 1024, 0, stream>>>(wsRows, F, Q, outMode);
}